// EdgePool_31825707664027
// MI455X (gfx1250) — compile-verified
//
#include <hip/hip_runtime.h>
#include <stdint.h>

#define Nn 30000
#define Ee 240000
#define Dd 64
#define INn 128
#define Gg 64
#define Cc 10
#define HASH_N (1u<<19)
#define HASH_MASK (HASH_N-1u)
#define NBLK 938            // ceil(Ee/256)

#define XSTRIDE 132         // 128 + 4 pad dwords (TDM pad_interval=128DW, pad_amount=4DW)
#define WSTRIDE 80          // 64 + 16 pad dwords (TDM pad_interval=64DW,  pad_amount=16DW)

typedef __attribute__((ext_vector_type(2))) float v2f;
typedef __attribute__((ext_vector_type(8))) float v8f;
typedef __attribute__((ext_vector_type(4))) uint32_t u32x4;
typedef __attribute__((ext_vector_type(8))) uint32_t u32x8;

// ---- monotone float <-> ordered-uint mapping (for atomicMax / radix keys) ----
__device__ __forceinline__ uint32_t ford(float f){
  uint32_t u = __float_as_uint(f);
  return (u & 0x80000000u) ? ~u : (u | 0x80000000u);
}
__device__ __forceinline__ float finv(uint32_t u){
  u = (u & 0x80000000u) ? (u ^ 0x80000000u) : ~u;
  return __uint_as_float(u);
}

// ------------- Tensor Data Mover: 2D tile global->LDS (2-group D#) -------------
// D# group0: [1:0]=count=1, [63:32]=lds_addr, [120:64]=global_addr, [127:126]=type=2
// D# group1: [17:16]=data_size(2=4B), [20]=pad_enable, [24:22]=pad_interval,
//            [31:25]=pad_amount, [79:48]=tensor_dim0, [111:80]=tensor_dim1,
//            [127:112]=tile_dim0, [143:128]=tile_dim1, [207:160]=tensor_dim0_stride
__device__ __forceinline__ void tdm_load_2d(uint32_t lds_off, const void* gptr,
                                            uint32_t tensor_d0, uint32_t tensor_d1,
                                            uint32_t tile_d0, uint32_t tile_d1,
                                            uint32_t stride_d0, uint32_t ctrl){
  uint64_t ga = (uint64_t)(uintptr_t)gptr;
  u32x4 g0 = { 1u, lds_off, (uint32_t)ga,
               (uint32_t)((ga >> 32) & 0x1FFFFFFu) | 0x80000000u /*type=2*/ };
  u32x8 g1 = { ctrl,
               (tensor_d0 & 0xFFFFu) << 16,
               (tensor_d0 >> 16) | ((tensor_d1 & 0xFFFFu) << 16),
               (tensor_d1 >> 16) | (tile_d0 << 16),
               tile_d1,
               stride_d0,
               0u, 0u };
  asm volatile("tensor_load_to_lds %0, %1" :: "s"(g0), "s"(g1) : "memory");
}
// ctrl = data_size=2<<16 | pad_enable<<20 | pad_interval<<22 | pad_amount<<25
#define TDM_CTRL_X ((2u<<16) | (1u<<20) | (6u<<22) | (3u<<25))   // 128DW + 4DW pad
#define TDM_CTRL_W ((2u<<16) | (1u<<20) | (5u<<22) | (15u<<25))  // 64DW + 16DW pad

// ------------------------------- fills -------------------------------
__global__ void k_fill_f32(float* p, float v, int n){
  int i = blockIdx.x*blockDim.x + threadIdx.x; if (i < n) p[i] = v;
}
__global__ void k_fill_u32(uint32_t* p, uint32_t v, int n){
  int i = blockIdx.x*blockDim.x + threadIdx.x; if (i < n) p[i] = v;
}
__global__ void k_fill_i32(int* p, int v, int n){
  int i = blockIdx.x*blockDim.x + threadIdx.x; if (i < n) p[i] = v;
}
__global__ void k_iota_i32(int* p, int n){
  int i = blockIdx.x*blockDim.x + threadIdx.x; if (i < n) p[i] = i;
}

// ------------------- node embedding GEMM via f32 WMMA + TDM -------------------
// z[N,64] = x[N,128] @ W[128,64].  One block = 16 rows, 4 waves = 4 col tiles.
// TDM stages both tiles into LDS with hardware padding that makes the A fetches
// (stride 132) and B fetches (row skew 16 banks) LDS-bank-conflict-free.
__global__ void k_embed(const float* __restrict__ x,
                        const float* __restrict__ W,
                        float* __restrict__ z){
  __shared__ float Wl[INn*WSTRIDE];   // 40 KB (padded 128x64)
  __shared__ float Xl[16*XSTRIDE];    // 8.25 KB (padded 16x128)

  const int wave = threadIdx.x >> 5;
  const int lane = threadIdx.x & 31;
  if (wave == 0){
    tdm_load_2d((uint32_t)(uintptr_t)&Xl[0], x + (size_t)blockIdx.x*16*INn,
                INn, Nn, INn, 16, INn, TDM_CTRL_X);
    __builtin_amdgcn_s_wait_tensorcnt(0);
  } else if (wave == 1){
    tdm_load_2d((uint32_t)(uintptr_t)&Wl[0], W,
                Dd, INn, Dd, INn, Dd, TDM_CTRL_W);
    __builtin_amdgcn_s_wait_tensorcnt(0);
  }
  __syncthreads();

  const int n0  = wave << 4;          // col tile
  const int m0  = blockIdx.x << 4;    // row tile
  const int mlo = lane & 15;
  const int hi  = lane >> 4;          // 0: K pair {k,k+1}; 1: {k+2,k+3}
  const int col = n0 + mlo;

  v8f c = {0.f,0.f,0.f,0.f,0.f,0.f,0.f,0.f};
  const float* xr = &Xl[mlo*XSTRIDE + hi*2];
  #pragma unroll
  for (int kk = 0; kk < INn; kk += 4){
    v2f a, b;
    a.x = xr[kk];
    a.y = xr[kk + 1];
    b.x = Wl[(kk + hi*2    )*WSTRIDE + col];
    b.y = Wl[(kk + hi*2 + 1)*WSTRIDE + col];
    c = __builtin_amdgcn_wmma_f32_16x16x4_f32(false, a, false, b,
                                              (short)0, c, false, false);
  }
  const int rbase = m0 + hi*8;
  #pragma unroll
  for (int r = 0; r < 8; ++r)
    z[(size_t)(rbase + r)*Dd + col] = c[r];
}

// -------- deterministic global_add_pool via binary search on sorted batch --------
__global__ void k_pool(const float* __restrict__ z, const int* __restrict__ batch,
                       float* __restrict__ emb){
  int i = blockIdx.x*blockDim.x + threadIdx.x;
  if (i >= Gg*Dd) return;
  int g = i >> 6, d = i & 63;
  int lo = 0, hi = Nn;
  while (lo < hi){ int m = (lo+hi)>>1; if (batch[m] <  g) lo = m+1; else hi = m; }
  int st = lo; lo = 0; hi = Nn;
  while (lo < hi){ int m = (lo+hi)>>1; if (batch[m] <= g) lo = m+1; else hi = m; }
  float acc = 0.f;
  for (int n = st; n < lo; ++n) acc += z[(size_t)n*Dd + d];
  emb[i] = acc;
}

// ----------------------- edge raw score (masked) -----------------------
__global__ void k_edge_score(const float* __restrict__ z, const int* __restrict__ src,
                             const int* __restrict__ dst, const uint32_t* __restrict__ em,
                             const float* __restrict__ We, const float* __restrict__ be,
                             int l, float* __restrict__ r, int e){
  __shared__ float w[2*Dd];
  if (threadIdx.x < 2*Dd) w[threadIdx.x] = We[l*2*Dd + threadIdx.x];
  __syncthreads();
  int i = blockIdx.x*blockDim.x + threadIdx.x;
  if (i >= e) return;
  const float4* zs = (const float4*)(z + (size_t)src[i]*Dd);
  const float4* zd = (const float4*)(z + (size_t)dst[i]*Dd);
  float acc = be[l];
  #pragma unroll
  for (int q = 0; q < Dd/4; ++q){
    float4 a = zs[q];
    acc += a.x*w[q*4] + a.y*w[q*4+1] + a.z*w[q*4+2] + a.w*w[q*4+3];
    float4 b = zd[q];
    acc += b.x*w[Dd+q*4] + b.y*w[Dd+q*4+1] + b.z*w[Dd+q*4+2] + b.w*w[Dd+q*4+3];
  }
  r[i] = em[i] ? acc : -1e30f;
}

// -------------------- segment softmax (by dst node) --------------------
__global__ void k_seg_max(const float* __restrict__ r, const int* __restrict__ dst,
                          uint32_t* __restrict__ mo, int e){
  int i = blockIdx.x*blockDim.x + threadIdx.x; if (i >= e) return;
  atomicMax(&mo[dst[i]], ford(r[i]));
}
__global__ void k_exp_den(const float* __restrict__ r, const int* __restrict__ dst,
                          const uint32_t* __restrict__ em, const uint32_t* __restrict__ mo,
                          float* __restrict__ den, float* __restrict__ exb, int e){
  int i = blockIdx.x*blockDim.x + threadIdx.x; if (i >= e) return;
  float ex = 0.f;
  if (em[i]){
    int t = dst[i];
    ex = expf(r[i] - finv(mo[t]));
    atomicAdd(&den[t], ex);
  }
  exb[i] = ex;
}
__global__ void k_score_key(const float* __restrict__ exb, const float* __restrict__ den,
                            const int* __restrict__ dst, const uint32_t* __restrict__ em,
                            float* __restrict__ eplus, uint32_t* __restrict__ key,
                            uint32_t* __restrict__ idx, int e){
  int i = blockIdx.x*blockDim.x + threadIdx.x; if (i >= e) return;
  float sc = exb[i] / fmaxf(den[dst[i]], 1e-16f);
  float ep = sc + 0.5f;
  eplus[i] = ep;
  key[i] = em[i] ? ford(-ep) : 0xFFFFFFFFu;   // ascending key for (-e_plus), masked last
  idx[i] = i;
}

// ------------------- stable LSD radix sort (8-bit digits) -------------------
__global__ void k_radix_hist(const uint32_t* __restrict__ key, uint32_t* __restrict__ hist,
                             int n, int shift){
  __shared__ uint32_t h[256];
  h[threadIdx.x] = 0; __syncthreads();
  int gid = blockIdx.x*256 + threadIdx.x;
  if (gid < n) atomicAdd(&h[(key[gid] >> shift) & 0xFFu], 1u);
  __syncthreads();
  hist[(uint32_t)threadIdx.x*NBLK + blockIdx.x] = h[threadIdx.x];
}
__global__ void k_radix_scan(uint32_t* __restrict__ hist, uint32_t* __restrict__ binstart){
  __shared__ uint32_t tot[256];
  __shared__ uint32_t bs[256];
  int t = threadIdx.x;
  uint32_t run = 0;
  for (int b = 0; b < NBLK; ++b){
    uint32_t c = hist[(uint32_t)t*NBLK + b];
    hist[(uint32_t)t*NBLK + b] = run;
    run += c;
  }
  tot[t] = run; __syncthreads();
  if (t == 0){ uint32_t acc = 0; for (int i = 0; i < 256; ++i){ bs[i] = acc; acc += tot[i]; } }
  __syncthreads();
  binstart[t] = bs[t];
}
__global__ void k_radix_scatter(const uint32_t* __restrict__ kin, const uint32_t* __restrict__ vin,
                                uint32_t* __restrict__ kout, uint32_t* __restrict__ vout,
                                const uint32_t* __restrict__ hist,
                                const uint32_t* __restrict__ binstart, int n, int shift){
  __shared__ uint32_t digs[256];
  int gid = blockIdx.x*256 + threadIdx.x;
  bool valid = gid < n;
  uint32_t k = 0, v = 0, dig = 0xFFFFFFFFu;
  if (valid){ k = kin[gid]; v = vin[gid]; dig = (k >> shift) & 0xFFu; }
  digs[threadIdx.x] = dig;
  __syncthreads();
  if (valid){
    uint32_t rank = 0;
    for (int j = 0; j < (int)threadIdx.x; ++j) rank += (digs[j] == dig);
    uint32_t pos = binstart[dig] + hist[dig*NBLK + blockIdx.x] + rank;
    kout[pos] = k; vout[pos] = v;
  }
}

// --------- pack sorted edges: {s|(t<<16), eid|(mask<<31)} (s,t < 32768) ---------
__global__ void k_gather_sorted(const uint32_t* __restrict__ order, const int* __restrict__ src,
                                const int* __restrict__ dst, const uint32_t* __restrict__ em,
                                uint2* __restrict__ packed, int e){
  int i = blockIdx.x*blockDim.x + threadIdx.x; if (i >= e) return;
  uint32_t eid = order[i];
  uint32_t s = (uint32_t)src[eid], t = (uint32_t)dst[eid];
  packed[i] = make_uint2(s | (t << 16), eid | (em[eid] ? 0x80000000u : 0u));
}

// -------- sequential greedy contraction; avail bitmask resident in LDS --------
__global__ void k_contract(const uint2* __restrict__ packed, const uint32_t* __restrict__ nmask,
                           int* __restrict__ cluster, int* __restrict__ picked){
  __shared__ uint32_t avail[(Nn + 31)/32];
  for (int i = threadIdx.x; i < (Nn + 31)/32; i += blockDim.x){
    uint32_t w = 0;
    for (int b = 0; b < 32; ++b){
      int n = i*32 + b;
      if (n < Nn && nmask[n]) w |= (1u << b);
    }
    avail[i] = w;
  }
  __syncthreads();
  if (threadIdx.x == 0){
    for (int i = 0; i < Ee; ++i){
      __builtin_prefetch(&packed[i + 64], 0, 0);
      uint2 p = packed[i];
      int s = (int)(p.x & 0xFFFFu);
      int t = (int)(p.x >> 16);
      uint32_t em = p.y >> 31;
      int eid = (int)(p.y & 0x7FFFFFFFu);
      uint32_t bs = (avail[s >> 5] >> (s & 31)) & 1u;
      uint32_t bt = (avail[t >> 5] >> (t & 31)) & 1u;
      if (em & bs & bt){
        avail[s >> 5] &= ~(1u << (s & 31));
        avail[t >> 5] &= ~(1u << (t & 31));
        cluster[t] = s;
        picked[s]  = eid;
      }
    }
  }
}

// --------------- cluster scatter-sum (<=2 contributors: deterministic) ---------------
__global__ void k_scatter_x(const float* __restrict__ z, const int* __restrict__ cluster,
                            float* __restrict__ nz, int n){
  int i = blockIdx.x*blockDim.x + threadIdx.x; if (i >= n) return;
  int node = i >> 6, d = i & 63;
  atomicAdd(&nz[((size_t)cluster[node] << 6) + d], z[i]);
}
__global__ void k_scatter_mask(const uint32_t* __restrict__ nmin, const int* __restrict__ cluster,
                               uint32_t* __restrict__ nmout, int n){
  int i = blockIdx.x*blockDim.x + threadIdx.x; if (i >= n) return;
  if (nmin[i]) atomicOr(&nmout[cluster[i]], 1u);
}
__global__ void k_scale(float* __restrict__ nz, const int* __restrict__ picked,
                        const float* __restrict__ eplus, int n){
  int i = blockIdx.x*blockDim.x + threadIdx.x; if (i >= n) return;
  int node = i >> 6;
  int p = picked[node];
  float m = (p < Ee) ? eplus[p] : 1.0f;
  nz[i] *= m;
}

// ------------- remap edges + hash-set dedup (equiv. to sort+first-keep) -------------
__global__ void k_remap(const int* __restrict__ src, const int* __restrict__ dst,
                        const uint32_t* __restrict__ em, const int* __restrict__ cluster,
                        int* __restrict__ osrc, int* __restrict__ odst,
                        uint32_t* __restrict__ oem, uint32_t* __restrict__ hash, int e){
  int i = blockIdx.x*blockDim.x + threadIdx.x; if (i >= e) return;
  int ns = cluster[src[i]], nt = cluster[dst[i]];
  osrc[i] = ns; odst[i] = nt;
  uint32_t keep = 0;
  if (em[i]){
    uint32_t key = (uint32_t)ns * (uint32_t)Nn + (uint32_t)nt;   // < 9e8, != sentinel
    uint32_t h = (key * 2654435761u) & HASH_MASK;
    for (;;){
      uint32_t old = atomicCAS(&hash[h], 0xFFFFFFFFu, key);
      if (old == 0xFFFFFFFFu){ keep = 1; break; }
      if (old == key){ keep = 0; break; }
      h = (h + 1) & HASH_MASK;
    }
  }
  oem[i] = keep;
}

// ----------------------------- final FC [64,192]x[192,10] -----------------------------
__global__ void k_final(const float* __restrict__ emb, const float* __restrict__ W_fc,
                        const float* __restrict__ b_fc, float* __restrict__ out){
  int i = blockIdx.x*blockDim.x + threadIdx.x;
  if (i >= Gg*Cc) return;
  int g = i / Cc, c = i % Cc;
  float acc = b_fc[c];
  #pragma unroll
  for (int l = 0; l < 3; ++l)
    for (int d = 0; d < Dd; ++d)
      acc += emb[(size_t)l*Gg*Dd + g*Dd + d] * W_fc[(l*Dd + d)*Cc + c];
  out[g*Cc + c] = acc;
}

// ======================================================================
extern "C" void kernel_launch(void* const* d_in, const int* in_sizes, int n_in,
                              void* d_out, int out_size, void* d_ws, size_t ws_size,
                              hipStream_t stream) {
  (void)in_sizes; (void)n_in; (void)out_size; (void)ws_size;
  const float* x       = (const float*)d_in[0];
  const int*   ei      = (const int*)  d_in[1];
  const int*   batch   = (const int*)  d_in[2];
  const float* W_embed = (const float*)d_in[3];
  const float* We      = (const float*)d_in[4];
  const float* be      = (const float*)d_in[5];
  const float* W_fc    = (const float*)d_in[6];
  const float* b_fc    = (const float*)d_in[7];
  float* out = (float*)d_out;

  char* wsp = (char*)d_ws;
  auto alloc = [&](size_t bytes)->void*{
    void* p = (void*)wsp; wsp += (bytes + 255) & ~(size_t)255; return p;
  };
  float*    zA      = (float*)   alloc(sizeof(float)*Nn*Dd);
  float*    zB      = (float*)   alloc(sizeof(float)*Nn*Dd);
  float*    rbuf    = (float*)   alloc(sizeof(float)*Ee);
  float*    exb     = (float*)   alloc(sizeof(float)*Ee);
  float*    eplus   = (float*)   alloc(sizeof(float)*Ee);
  uint32_t* key     = (uint32_t*)alloc(4*(size_t)Ee);
  uint32_t* idxb    = (uint32_t*)alloc(4*(size_t)Ee);
  uint32_t* key2    = (uint32_t*)alloc(4*(size_t)Ee);
  uint32_t* idx2    = (uint32_t*)alloc(4*(size_t)Ee);
  uint2*    packed  = (uint2*)   alloc(8*(size_t)Ee);
  uint32_t* mo      = (uint32_t*)alloc(4*(size_t)Nn);
  float*    den     = (float*)   alloc(4*(size_t)Nn);
  int*      cluster = (int*)     alloc(4*(size_t)Nn);
  int*      picked  = (int*)     alloc(4*(size_t)Nn);
  uint32_t* nmA     = (uint32_t*)alloc(4*(size_t)Nn);
  uint32_t* nmB     = (uint32_t*)alloc(4*(size_t)Nn);
  int*      srcA    = (int*)     alloc(4*(size_t)Ee);
  int*      dstA    = (int*)     alloc(4*(size_t)Ee);
  int*      srcB    = (int*)     alloc(4*(size_t)Ee);
  int*      dstB    = (int*)     alloc(4*(size_t)Ee);
  uint32_t* emA     = (uint32_t*)alloc(4*(size_t)Ee);
  uint32_t* emB     = (uint32_t*)alloc(4*(size_t)Ee);
  uint32_t* hash    = (uint32_t*)alloc(4*(size_t)HASH_N);
  uint32_t* hist    = (uint32_t*)alloc(4*(size_t)256*NBLK);
  uint32_t* binst   = (uint32_t*)alloc(4*256);
  float*    emb     = (float*)   alloc(4*(size_t)3*Gg*Dd);

  const int EB  = (Ee + 255)/256;
  const int NBk = (Nn + 255)/256;
  const int NDB = (Nn*Dd + 255)/256;
  const int GDB = (Gg*Dd + 255)/256;

  // ---- embed + pool level 0 ----
  k_embed<<<Nn/16, 128, 0, stream>>>(x, W_embed, zA);
  k_pool <<<GDB, 256, 0, stream>>>(zA, batch, emb);
  k_fill_u32<<<NBk, 256, 0, stream>>>(nmA, 1u, Nn);
  k_fill_u32<<<EB,  256, 0, stream>>>(emA, 1u, Ee);

  auto layer = [&](const float* z, float* nz,
                   const int* src, const int* dst, const uint32_t* em,
                   const uint32_t* nmin,
                   int* osrc, int* odst, uint32_t* oem, uint32_t* nmout, int l){
    k_fill_u32<<<NBk, 256, 0, stream>>>(mo, 0u, Nn);
    k_fill_f32<<<NBk, 256, 0, stream>>>(den, 0.f, Nn);
    k_edge_score<<<EB, 256, 0, stream>>>(z, src, dst, em, We, be, l, rbuf, Ee);
    k_seg_max  <<<EB, 256, 0, stream>>>(rbuf, dst, mo, Ee);
    k_exp_den  <<<EB, 256, 0, stream>>>(rbuf, dst, em, mo, den, exb, Ee);
    k_score_key<<<EB, 256, 0, stream>>>(exb, den, dst, em, eplus, key, idxb, Ee);
    uint32_t *kin = key, *vin = idxb, *kout = key2, *vout = idx2;
    for (int p = 0; p < 4; ++p){
      k_radix_hist   <<<NBLK, 256, 0, stream>>>(kin, hist, Ee, p*8);
      k_radix_scan   <<<1,    256, 0, stream>>>(hist, binst);
      k_radix_scatter<<<NBLK, 256, 0, stream>>>(kin, vin, kout, vout, hist, binst, Ee, p*8);
      uint32_t* t = kin; kin = kout; kout = t;
      t = vin; vin = vout; vout = t;
    }
    k_gather_sorted<<<EB, 256, 0, stream>>>(vin, src, dst, em, packed, Ee);
    k_iota_i32<<<NBk, 256, 0, stream>>>(cluster, Nn);
    k_fill_i32<<<NBk, 256, 0, stream>>>(picked, Ee, Nn);
    k_contract<<<1, 128, 0, stream>>>(packed, nmin, cluster, picked);
    k_fill_f32<<<NDB, 256, 0, stream>>>(nz, 0.f, Nn*Dd);
    k_fill_u32<<<NBk, 256, 0, stream>>>(nmout, 0u, Nn);
    k_scatter_x   <<<NDB, 256, 0, stream>>>(z, cluster, nz, Nn*Dd);
    k_scatter_mask<<<NBk, 256, 0, stream>>>(nmin, cluster, nmout, Nn);
    k_scale       <<<NDB, 256, 0, stream>>>(nz, picked, eplus, Nn*Dd);
    k_fill_u32<<<(HASH_N + 255)/256, 256, 0, stream>>>(hash, 0xFFFFFFFFu, HASH_N);
    k_remap<<<EB, 256, 0, stream>>>(src, dst, em, cluster, osrc, odst, oem, hash, Ee);
    k_pool<<<GDB, 256, 0, stream>>>(nz, batch, emb + (size_t)(l + 1)*Gg*Dd);
  };

  layer(zA, zB, ei,   ei + Ee, emA, nmA, srcA, dstA, emB, nmB, 0);
  layer(zB, zA, srcA, dstA,    emB, nmB, srcB, dstB, emA, nmA, 1);

  k_final<<<(Gg*Cc + 255)/256, 256, 0, stream>>>(emb, W_fc, b_fc, out);
}